// GlobalGraphEncoder_8985071583974
// MI455X (gfx1250) — compile-verified
//
#include <hip/hip_runtime.h>
#include <math.h>

typedef __attribute__((ext_vector_type(2))) float v2f;
typedef __attribute__((ext_vector_type(8))) float v8f;

#define HID 128

// GEMM modes (compile-time -> branchless epilogue)
#define MODE_BIAS_RELU 0   // input projection:  C = relu(A@W + bias)
#define MODE_AGG       1   // conv layer:        C = A@W; agg = C * selfnorm[r]
#define MODE_BIAS      2   // output projection: C = A@W + bias

// ---------------------------------------------------------------------------
// C[nrows,128] = A[nrows,128] @ W[128,128], 256-thread block = 8 waves,
// block computes a 128x128 tile; each wave owns a 16-row strip with 8 16x16
// f32 accumulators -> 32 K-steps x 8 tiles of v_wmma_f32_16x16x4_f32.
// W is staged in LDS pair-interleaved: element (k,col) at ((k>>1)*128+col)*2
// + (k&1), so each B fragment {W[k][col], W[k+1][col]} is one aligned
// ds_load_b64 straight into an even VGPR pair (no mov shuffling).
// ---------------------------------------------------------------------------
template <int MODE>
__global__ __launch_bounds__(256) void gemm128_wmma(
    const float* __restrict__ A,
    const float* __restrict__ W,
    const float* __restrict__ bias,      // MODE 0/2
    float* __restrict__ C,
    float* __restrict__ agg,             // MODE 1
    const float* __restrict__ selfnorm,  // MODE 1
    int nrows)
{
    __shared__ __align__(16) float Ws[HID * HID];  // 64 KB of 320 KB/WGP
    const int tid = threadIdx.x;
    for (int idx = tid; idx < HID * HID; idx += 256) {
        const int k   = idx >> 7;
        const int col = idx & 127;
        Ws[(((k >> 1) * HID) + col) * 2 + (k & 1)] = W[idx];
    }
    __syncthreads();

    const int wave  = tid >> 5;
    const int lane  = tid & 31;
    const int m     = lane & 15;   // row-in-tile (A) / col-in-tile (B,C)
    const int hi    = lane >> 4;   // upper half-wave
    const int wrow0 = blockIdx.x * 128 + wave * 16;

    const int arown   = wrow0 + m;
    const int rclamp  = arown < nrows ? arown : (nrows - 1);
    const float* arow = A + (size_t)rclamp * HID;
    const v2f* Wp     = (const v2f*)Ws;   // pair index = (k>>1)*128 + col

    v8f acc[8] = {};

    for (int k0 = 0; k0 < HID; k0 += 4) {
        const int k  = k0 + 2 * hi;        // ISA 16x4 f32 A-fragment layout
        const int kp = (k0 >> 1) + hi;     // pair row in interleaved LDS
        v2f a = *(const v2f*)&arow[k];     // global_load_b64
#pragma unroll
        for (int ct = 0; ct < 8; ct++) {
            v2f b = Wp[kp * HID + ct * 16 + m];   // ds_load_b64, aligned pair
            acc[ct] = __builtin_amdgcn_wmma_f32_16x16x4_f32(
                false, a, false, b, (short)0, acc[ct], false, false);
        }
    }

    float bv[8];
#pragma unroll
    for (int ct = 0; ct < 8; ct++)
        bv[ct] = (MODE != MODE_AGG) ? bias[ct * 16 + m] : 0.f;

#pragma unroll
    for (int v = 0; v < 8; v++) {
        const int r = wrow0 + v + 8 * hi;  // ISA 16x16 f32 C/D layout
        if (r < nrows) {
            const size_t base = (size_t)r * HID + m;
            const float sn = (MODE == MODE_AGG) ? selfnorm[r] : 0.f;
#pragma unroll
            for (int ct = 0; ct < 8; ct++) {
                float val = acc[ct][v] + bv[ct];
                if (MODE == MODE_BIAS_RELU) val = fmaxf(val, 0.f);
                C[base + ct * 16] = val;
                if (MODE == MODE_AGG) agg[base + ct * 16] = val * sn;
            }
        }
    }
}

// ---------------------------------------------------------------------------
// ew[e] = sigmoid( relu(edge_attr[e] @ We1 + be1) @ We2 + be2 )
// One thread per edge; We1 (12 KB) staged in LDS (broadcast reads).
// ---------------------------------------------------------------------------
__global__ __launch_bounds__(256) void edge_mlp_kernel(
    const float* __restrict__ edge_attr,
    const float* __restrict__ We1, const float* __restrict__ be1,
    const float* __restrict__ We2, const float* __restrict__ be2,
    float* __restrict__ ew, int E)
{
    __shared__ float W1s[32 * 96];
    __shared__ float b1s[96];
    __shared__ float W2s[96];
    const int tid = threadIdx.x;
    for (int i = tid; i < 32 * 96; i += 256) W1s[i] = We1[i];
    if (tid < 96) { b1s[tid] = be1[tid]; W2s[tid] = We2[tid]; }
    __syncthreads();

    const int e = blockIdx.x * 256 + tid;
    if (e >= E) return;

    float a[32];
    const float4* ap = (const float4*)(edge_attr + (size_t)e * 32);
#pragma unroll
    for (int i = 0; i < 8; i++) {
        float4 t = ap[i];
        a[4*i] = t.x; a[4*i+1] = t.y; a[4*i+2] = t.z; a[4*i+3] = t.w;
    }
    float acc = be2[0];
    for (int j = 0; j < 96; j++) {
        float s = b1s[j];
#pragma unroll
        for (int k = 0; k < 32; k++) s = fmaf(a[k], W1s[k * 96 + j], s);
        s = fmaxf(s, 0.f);
        acc = fmaf(s, W2s[j], acc);
    }
    ew[e] = 1.f / (1.f + __expf(-acc));
}

__global__ __launch_bounds__(256) void fill1_kernel(float* __restrict__ p, int n) {
    int i = blockIdx.x * 256 + threadIdx.x;
    if (i < n) p[i] = 1.0f;  // self-loop weight
}

__global__ __launch_bounds__(256) void deg_scatter_kernel(
    const int* __restrict__ dst, const float* __restrict__ ew,
    float* __restrict__ deg, int E)
{
    int e = blockIdx.x * 256 + threadIdx.x;
    if (e < E) atomicAdd(&deg[dst[e]], ew[e]);
}

__global__ __launch_bounds__(256) void dis_kernel(
    const float* __restrict__ deg, float* __restrict__ dis,
    float* __restrict__ selfnorm, int n)
{
    int i = blockIdx.x * 256 + threadIdx.x;
    if (i < n) {
        float d = deg[i];                 // >= 1 always (self-loop)
        float r = 1.0f / sqrtf(d);
        dis[i] = r;
        selfnorm[i] = r * r;              // dis[i]*1*dis[i]
    }
}

__global__ __launch_bounds__(256) void norm_kernel(
    const int* __restrict__ src, const int* __restrict__ dst,
    const float* __restrict__ ew, const float* __restrict__ dis,
    float* __restrict__ nrm, int E)
{
    int e = blockIdx.x * 256 + threadIdx.x;
    if (e < E) nrm[e] = dis[src[e]] * ew[e] * dis[dst[e]];
}

// agg[dst[e], c] += hW[src[e], c] * norm[e]; 128 consecutive lanes per edge so
// gathers/atomics are 128B-coalesced; hW/agg (25.6 MB each) are L2-resident
// in the 192 MB L2, so the random node accesses never touch HBM.
__global__ __launch_bounds__(256) void scatter_kernel(
    const float* __restrict__ hW,
    const int* __restrict__ src, const int* __restrict__ dst,
    const float* __restrict__ nrm,
    float* __restrict__ agg, int E)
{
    size_t gid = (size_t)blockIdx.x * 256 + threadIdx.x;
    if (gid >= (size_t)E * 128) return;
    int e = (int)(gid >> 7);
    int c = (int)(gid & 127);
    float v = hW[(size_t)src[e] * 128 + c] * nrm[e];
    atomicAdd(&agg[(size_t)dst[e] * 128 + c], v);
}

// h = relu(agg + bc + h)   (residual fused)
__global__ __launch_bounds__(256) void finalize_kernel(
    const float* __restrict__ agg, const float* __restrict__ bc_l,
    float* __restrict__ h, int total)
{
    int i = blockIdx.x * 256 + threadIdx.x;
    if (i < total) {
        int c = i & 127;
        h[i] = fmaxf(agg[i] + bc_l[c] + h[i], 0.f);
    }
}

extern "C" void kernel_launch(void* const* d_in, const int* in_sizes, int n_in,
                              void* d_out, int out_size, void* d_ws, size_t ws_size,
                              hipStream_t stream)
{
    const float* x         = (const float*)d_in[0];
    const int*   edge_idx  = (const int*)  d_in[1];
    const float* edge_attr = (const float*)d_in[2];
    const float* W_in      = (const float*)d_in[3];
    const float* b_in      = (const float*)d_in[4];
    const float* We1       = (const float*)d_in[5];
    const float* be1       = (const float*)d_in[6];
    const float* We2       = (const float*)d_in[7];
    const float* be2       = (const float*)d_in[8];
    const float* Wc        = (const float*)d_in[9];
    const float* bc        = (const float*)d_in[10];
    const float* W_out     = (const float*)d_in[11];
    const float* b_out     = (const float*)d_in[12];

    const int N = in_sizes[0] / 128;
    const int E = in_sizes[1] / 2;
    const int* src = edge_idx;       // edge_index[0]
    const int* dst = edge_idx + E;   // edge_index[1]

    float* w = (float*)d_ws;
    float* h   = w; w += (size_t)N * 128;
    float* hW  = w; w += (size_t)N * 128;
    float* agg = w; w += (size_t)N * 128;
    float* ew  = w; w += E;
    float* nrm = w; w += E;
    float* deg = w; w += N;
    float* dis = w; w += N;
    float* sn  = w; w += N;

    const int tb = 256;
    const int eb = (E + tb - 1) / tb;
    const int nb = (N + tb - 1) / tb;

    // Edge weights + GCN normalization (shared across layers)
    edge_mlp_kernel<<<eb, tb, 0, stream>>>(edge_attr, We1, be1, We2, be2, ew, E);
    fill1_kernel<<<nb, tb, 0, stream>>>(deg, N);
    deg_scatter_kernel<<<eb, tb, 0, stream>>>(dst, ew, deg, E);
    dis_kernel<<<nb, tb, 0, stream>>>(deg, dis, sn, N);
    norm_kernel<<<eb, tb, 0, stream>>>(src, dst, ew, dis, nrm, E);

    const int gblocks = (N + 127) / 128;
    // h = relu(x @ W_in + b_in)
    gemm128_wmma<MODE_BIAS_RELU><<<gblocks, 256, 0, stream>>>(
        x, W_in, b_in, h, nullptr, nullptr, N);

    const int total   = N * 128;
    const int fblocks = (total + tb - 1) / tb;
    const size_t sc   = (size_t)E * 128;
    const int sblocks = (int)((sc + tb - 1) / tb);

    for (int l = 0; l < 4; l++) {
        // hW = h @ Wc[l];  agg initialized with self-loop term hW * (1/deg)
        gemm128_wmma<MODE_AGG><<<gblocks, 256, 0, stream>>>(
            h, Wc + (size_t)l * 128 * 128, nullptr, hW, agg, sn, N);
        scatter_kernel<<<sblocks, tb, 0, stream>>>(hW, src, dst, nrm, agg, E);
        finalize_kernel<<<fblocks, tb, 0, stream>>>(agg, bc + l * 128, h, total);
    }

    // out = h @ W_out + b_out
    gemm128_wmma<MODE_BIAS><<<gblocks, 256, 0, stream>>>(
        h, W_out, b_out, (float*)d_out, nullptr, nullptr, N);
}